// INT4AsymmetricWeightsDecompressor_7928509629313
// MI455X (gfx1250) — compile-verified
//
#include <hip/hip_runtime.h>

// INT4 asymmetric dequant: out[o][g*128 + 2i]   = ((packed[o][g][i] & 15) - zp[o][g]) * scale[o][g]
//                          out[o][g*128 + 2i+1] = ((packed[o][g][i] >> 4) - zp[o][g]) * scale[o][g]
// Pure streaming, HBM-bandwidth bound (~406 MB total traffic -> ~17.4 us floor @ 23.3 TB/s).
// Strategy: b128 non-temporal loads/stores (gfx1250 TH=NT cache policy), 8 packed int32
// (32 B) in -> 16 floats (64 B) out per thread, wave32-friendly 256-thread blocks.

typedef int   i32x4 __attribute__((ext_vector_type(4)));
typedef float f32x4 __attribute__((ext_vector_type(4)));

__global__ __launch_bounds__(256) void int4_asym_dequant_kernel(
    const int*   __restrict__ packed,      // (OUT, N_GROUPS, 64) int32 carrier, one byte each
    const float* __restrict__ scale,       // (OUT, N_GROUPS, 1)
    const float* __restrict__ zero_point,  // (OUT, N_GROUPS, 1)
    float*       __restrict__ out,         // (OUT, OUT) fp32
    int n_packed)                          // total packed elements (OUT*N_GROUPS*64)
{
    const int t = blockIdx.x * 256 + threadIdx.x;
    const long pbase = (long)t * 8;
    if (pbase >= (long)n_packed) return;

    // Two 128-bit non-temporal loads: 8 packed bytes (int32 carriers), read-once data.
    const i32x4* p = (const i32x4*)(packed + pbase);
    i32x4 a = __builtin_nontemporal_load(p);
    i32x4 b = __builtin_nontemporal_load(p + 1);

    // Per-group quant params: 8 consecutive threads share one group -> cached, coalesced.
    const int g = t >> 3;
    const float s = scale[g];
    const float z = zero_point[g];

    // Dequant, matching reference rounding: (float(q) - z) * s  (q, z exact in fp32).
    f32x4 o0, o1, o2, o3;
    o0.x = ((float)(a.x & 15) - z) * s;  o0.y = ((float)(a.x >> 4) - z) * s;
    o0.z = ((float)(a.y & 15) - z) * s;  o0.w = ((float)(a.y >> 4) - z) * s;
    o1.x = ((float)(a.z & 15) - z) * s;  o1.y = ((float)(a.z >> 4) - z) * s;
    o1.z = ((float)(a.w & 15) - z) * s;  o1.w = ((float)(a.w >> 4) - z) * s;
    o2.x = ((float)(b.x & 15) - z) * s;  o2.y = ((float)(b.x >> 4) - z) * s;
    o2.z = ((float)(b.y & 15) - z) * s;  o2.w = ((float)(b.y >> 4) - z) * s;
    o3.x = ((float)(b.z & 15) - z) * s;  o3.y = ((float)(b.z >> 4) - z) * s;
    o3.z = ((float)(b.w & 15) - z) * s;  o3.w = ((float)(b.w >> 4) - z) * s;

    // Four 128-bit non-temporal stores: write-once 268 MB stream, keep it out of L2.
    f32x4* ob = (f32x4*)(out + (long)t * 16);
    __builtin_nontemporal_store(o0, ob + 0);
    __builtin_nontemporal_store(o1, ob + 1);
    __builtin_nontemporal_store(o2, ob + 2);
    __builtin_nontemporal_store(o3, ob + 3);
}

extern "C" void kernel_launch(void* const* d_in, const int* in_sizes, int n_in,
                              void* d_out, int out_size, void* d_ws, size_t ws_size,
                              hipStream_t stream) {
    const int*   packed     = (const int*)d_in[0];
    const float* scale      = (const float*)d_in[1];
    const float* zero_point = (const float*)d_in[2];
    float*       out        = (float*)d_out;

    const int n_packed = in_sizes[0];          // 8192*64*64 = 33,554,432
    const int threads  = (n_packed + 7) / 8;   // 8 packed int32 per thread
    const int blocks   = (threads + 255) / 256;

    int4_asym_dequant_kernel<<<blocks, 256, 0, stream>>>(packed, scale, zero_point, out, n_packed);
}